// GraphNeuralNetwork_49855980372316
// MI455X (gfx1250) — compile-verified
//
#include <hip/hip_runtime.h>

// ---------------------------------------------------------------------------
// 2-layer GCN for MI455X (gfx1250, wave32).
// Edge scatter/gather dominates (L2-resident random traffic); GEMMs use
// V_WMMA_F32_16X16X4_F32 to keep full f32 precision (GEMM FLOPs negligible).
// ---------------------------------------------------------------------------

typedef __attribute__((ext_vector_type(2))) float v2f;
typedef __attribute__((ext_vector_type(8))) float v8f;

#define NNODES 100000
#define NEDGES 1600000
#define DIM_IN 128
#define DIM_H  64
#define DIM_OUT 32
#define NLAYERS 2

// ------------------------------- utility -----------------------------------

// n must be a multiple of 4 (all our buffers are).
__global__ void gcn_zero4_kernel(float4* __restrict__ p, int n4) {
    int i = blockIdx.x * blockDim.x + threadIdx.x;
    if (i < n4) p[i] = make_float4(0.f, 0.f, 0.f, 0.f);
}

// deg[u] = #edges with dst == u
__global__ void gcn_deg_kernel(const int* __restrict__ dst,
                               float* __restrict__ deg, int E) {
    int e = blockIdx.x * blockDim.x + threadIdx.x;
    if (e < E) atomicAdd(&deg[dst[e]], 1.0f);
}

// inv_norm[e] = 1 / sqrt((deg[dst]+1)(deg[src]+1))
__global__ void gcn_invnorm_kernel(const int* __restrict__ dst,
                                   const int* __restrict__ src,
                                   const float* __restrict__ deg,
                                   float* __restrict__ inv_norm, int E) {
    int e = blockIdx.x * blockDim.x + threadIdx.x;
    if (e < E) {
        float d = (deg[dst[e]] + 1.0f) * (deg[src[e]] + 1.0f);
        inv_norm[e] = rsqrtf(d);
    }
}

// agg[dst[e]][0:64] += h[src[e]][0:64] * inv_norm[e]
// 16 consecutive threads = one edge; each thread owns a float4 feature chunk.
// Gather is a coalesced 256B row (global_load_b128 per lane); atomics are
// 4x global_atomic_add_f32 per lane into distinct addresses (L2-resident).
__global__ void gcn_scatter_kernel(const int* __restrict__ dst,
                                   const int* __restrict__ src,
                                   const float* __restrict__ inv_norm,
                                   const float* __restrict__ h,
                                   float* __restrict__ agg, int E) {
    unsigned tid = blockIdx.x * blockDim.x + threadIdx.x;
    unsigned e = tid >> 4;              // edge index (16 lanes per edge)
    unsigned f = (tid & 15u) << 2;      // feature offset: 0,4,...,60
    if (e < (unsigned)E) {
        int s = src[e];
        int d = dst[e];
        float w = inv_norm[e];
        const float4 hv = *(const float4*)(h + (size_t)s * DIM_H + f);
        float* ap = agg + (size_t)d * DIM_H + f;
        atomicAdd(ap + 0, hv.x * w);
        atomicAdd(ap + 1, hv.y * w);
        atomicAdd(ap + 2, hv.z * w);
        atomicAdd(ap + 3, hv.w * w);
    }
}

// ------------------------------ WMMA GEMM ----------------------------------
// C[M x NCOL] = op(A [+ A2]) @ B[K x NCOL], optional ReLU on output.
// One wave (32 threads) computes one 16x16 tile; K-loop in steps of 4 using
// V_WMMA_F32_16X16X4_F32.  M must be a multiple of 16 (100000 = 6250*16).
//
// ISA VGPR layouts (CDNA5 ISA):
//   A 16x4 : lanes 0-15 -> row M=lane, {K=k,k+1}; lanes 16-31 -> {K=k+2,k+3}
//   B 4x16 : lanes 0-15 -> col N=lane, {K=k,k+1}; lanes 16-31 -> {K=k+2,k+3}
//   C/D    : vgpr j -> M=j (lanes 0-15) / M=j+8 (lanes 16-31), N=lane%16
template <int K, int NCOL, bool RELU, bool ADD2>
__global__ void gcn_wmma_gemm_kernel(const float* __restrict__ A,
                                     const float* __restrict__ A2,
                                     const float* __restrict__ B,
                                     float* __restrict__ C) {
    const int lane = threadIdx.x;          // blockDim.x == 32
    const int half = lane >> 4;            // 0: K lo pair, 1: K hi pair
    const int l15  = lane & 15;

    const int row  = blockIdx.x * 16 + l15;   // A row this lane loads
    const int col  = blockIdx.y * 16 + l15;   // B/D column this lane owns

    v8f acc = {};

    #pragma unroll 4
    for (int k = 0; k < K; k += 4) {
        const int ka = k + half * 2;

        v2f a, b;
        float a0 = A[(size_t)row * K + ka];
        float a1 = A[(size_t)row * K + ka + 1];
        if (ADD2) {
            a0 += A2[(size_t)row * K + ka];
            a1 += A2[(size_t)row * K + ka + 1];
        }
        a.x = a0;
        a.y = a1;

        b.x = B[(size_t)(ka + 0) * NCOL + col];
        b.y = B[(size_t)(ka + 1) * NCOL + col];

        // 8 args: (neg_a, A, neg_b, B, c_mod, C, reuse_a, reuse_b)
        acc = __builtin_amdgcn_wmma_f32_16x16x4_f32(
            false, a, false, b, (short)0, acc, false, false);
    }

    const int row_base = blockIdx.x * 16 + half * 8;
    #pragma unroll
    for (int j = 0; j < 8; ++j) {
        float v = acc[j];
        if (RELU) v = v > 0.0f ? v : 0.0f;
        C[(size_t)(row_base + j) * NCOL + col] = v;
    }
}

// ------------------------------- driver ------------------------------------

static inline size_t align256(size_t x) { return (x + 255) & ~(size_t)255; }

extern "C" void kernel_launch(void* const* d_in, const int* in_sizes, int n_in,
                              void* d_out, int out_size, void* d_ws, size_t ws_size,
                              hipStream_t stream) {
    (void)in_sizes; (void)n_in; (void)out_size; (void)ws_size;

    const float* x        = (const float*)d_in[0];              // [N,128]
    const int*   ei       = (const int*)d_in[1];                // [2,E]
    const float* W_in     = (const float*)d_in[2];              // [128,64]
    const float* W_layers = (const float*)d_in[3];              // [2,64,64]
    const float* W_out    = (const float*)d_in[4];              // [64,32]
    float*       out      = (float*)d_out;                      // [N,32]

    const int* dst = ei;            // edge_index[0] : receiving node
    const int* src = ei + NEDGES;   // edge_index[1] : gathered neighbor

    // workspace carve-out (~84 MB)
    char* ws = (char*)d_ws;
    float* deg      = (float*)ws; ws += align256(sizeof(float) * NNODES);
    float* inv_norm = (float*)ws; ws += align256(sizeof(float) * NEDGES);
    float* hA       = (float*)ws; ws += align256(sizeof(float) * (size_t)NNODES * DIM_H);
    float* hB       = (float*)ws; ws += align256(sizeof(float) * (size_t)NNODES * DIM_H);
    float* agg      = (float*)ws; ws += align256(sizeof(float) * (size_t)NNODES * DIM_H);

    const int ROW_TILES = NNODES / 16;   // 6250

    // --- degrees + symmetric normalizer -------------------------------------
    gcn_zero4_kernel<<<(NNODES / 4 + 255) / 256, 256, 0, stream>>>(
        (float4*)deg, NNODES / 4);
    gcn_deg_kernel<<<(NEDGES + 255) / 256, 256, 0, stream>>>(dst, deg, NEDGES);
    gcn_invnorm_kernel<<<(NEDGES + 255) / 256, 256, 0, stream>>>(dst, src, deg,
                                                                 inv_norm, NEDGES);

    // --- h = x @ W_in  (K=128, NCOL=64) -------------------------------------
    gcn_wmma_gemm_kernel<DIM_IN, DIM_H, false, false>
        <<<dim3(ROW_TILES, DIM_H / 16), 32, 0, stream>>>(x, nullptr, W_in, hA);

    float* h_cur = hA;
    float* h_nxt = hB;

    // --- message-passing layers ---------------------------------------------
    const int aggN4 = NNODES * DIM_H / 4;                      // 1.6M float4
    const unsigned scatter_threads = (unsigned)NEDGES * 16u;   // 25.6M
    for (int l = 0; l < NLAYERS; ++l) {
        gcn_zero4_kernel<<<(aggN4 + 255) / 256, 256, 0, stream>>>(
            (float4*)agg, aggN4);
        gcn_scatter_kernel<<<(scatter_threads + 255u) / 256u, 256, 0, stream>>>(
            dst, src, inv_norm, h_cur, agg, NEDGES);
        // h = relu((agg + h) @ W_layers[l])
        gcn_wmma_gemm_kernel<DIM_H, DIM_H, true, true>
            <<<dim3(ROW_TILES, DIM_H / 16), 32, 0, stream>>>(
                agg, h_cur, W_layers + (size_t)l * DIM_H * DIM_H, h_nxt);
        float* t = h_cur; h_cur = h_nxt; h_nxt = t;
    }

    // --- out = h @ W_out (K=64, NCOL=32) ------------------------------------
    gcn_wmma_gemm_kernel<DIM_H, DIM_OUT, false, false>
        <<<dim3(ROW_TILES, DIM_OUT / 16), 32, 0, stream>>>(h_cur, nullptr, W_out, out);
}